// SRenderY_7370163879960
// MI455X (gfx1250) — compile-verified
//
#include <hip/hip_runtime.h>
#include <math.h>

// ---------------------------------------------------------------------------
// SRenderY fused renderer for MI455X (gfx1250, wave32).
//   Pass 1: zero vertex-normal accumulator (workspace)
//   Pass 2: face cross products -> atomicAdd into vertex normals
//   Pass 3: normalize vertex normals
//   Pass 4: fused per-pixel gather + barycentric interp + WMMA lighting
// Lighting uses V_WMMA_F32_16X16X4_F32: one 16-pixel tile per wave,
// lanes 16..31 mirror pixels 0..15 so A/B operand layouts need no shuffles
// (except 2 shfl's to feed ndl^T rows 2/3 into the second matmul's B half).
// ---------------------------------------------------------------------------

typedef __attribute__((ext_vector_type(2))) float v2f;
typedef __attribute__((ext_vector_type(8))) float v8f;

constexpr int Bc = 8;
constexpr int Vc = 5023;
constexpr int Fc = 9976;
constexpr int Hc = 512;
constexpr int Wc = 512;
constexpr int Lc = 5;
constexpr int HW = Hc * Wc;          // 262144, multiple of 16 -> waves never straddle a batch

// ---------------- Pass 1: zero workspace ----------------
__global__ void k_zero(float* __restrict__ p, int n) {
    int i = blockIdx.x * blockDim.x + threadIdx.x;
    if (i < n) p[i] = 0.0f;
}

// ---------------- Pass 2: face normal scatter-add ----------------
__global__ void k_face_accum(const float* __restrict__ verts,
                             const int* __restrict__ faces,
                             float* __restrict__ nrm) {
    int t = blockIdx.x * blockDim.x + threadIdx.x;
    if (t >= Bc * Fc) return;
    int b = t / Fc;
    int f = t - b * Fc;
    int i0 = faces[f * 3 + 0];
    int i1 = faces[f * 3 + 1];
    int i2 = faces[f * 3 + 2];
    const float* vb = verts + (size_t)b * Vc * 3;
    float ax = vb[i0 * 3 + 0], ay = vb[i0 * 3 + 1], az = vb[i0 * 3 + 2];
    float bx = vb[i1 * 3 + 0], by = vb[i1 * 3 + 1], bz = vb[i1 * 3 + 2];
    float cx = vb[i2 * 3 + 0], cy = vb[i2 * 3 + 1], cz = vb[i2 * 3 + 2];

    // reference: n[faces[:,1]] += cross(v2-v1, v0-v1)
    float e0x = cx - bx, e0y = cy - by, e0z = cz - bz;
    float e1x = ax - bx, e1y = ay - by, e1z = az - bz;
    float n1x = e0y * e1z - e0z * e1y;
    float n1y = e0z * e1x - e0x * e1z;
    float n1z = e0x * e1y - e0y * e1x;
    // n[faces[:,2]] += cross(v0-v2, v1-v2)
    float f0x = ax - cx, f0y = ay - cy, f0z = az - cz;
    float f1x = bx - cx, f1y = by - cy, f1z = bz - cz;
    float n2x = f0y * f1z - f0z * f1y;
    float n2y = f0z * f1x - f0x * f1z;
    float n2z = f0x * f1y - f0y * f1x;
    // n[faces[:,0]] += cross(v1-v0, v2-v0)
    float g0x = bx - ax, g0y = by - ay, g0z = bz - az;
    float g1x = cx - ax, g1y = cy - ay, g1z = cz - az;
    float n0x = g0y * g1z - g0z * g1y;
    float n0y = g0z * g1x - g0x * g1z;
    float n0z = g0x * g1y - g0y * g1x;

    float* nb = nrm + (size_t)b * Vc * 3;
    atomicAdd(&nb[i1 * 3 + 0], n1x);
    atomicAdd(&nb[i1 * 3 + 1], n1y);
    atomicAdd(&nb[i1 * 3 + 2], n1z);
    atomicAdd(&nb[i2 * 3 + 0], n2x);
    atomicAdd(&nb[i2 * 3 + 1], n2y);
    atomicAdd(&nb[i2 * 3 + 2], n2z);
    atomicAdd(&nb[i0 * 3 + 0], n0x);
    atomicAdd(&nb[i0 * 3 + 1], n0y);
    atomicAdd(&nb[i0 * 3 + 2], n0z);
}

// ---------------- Pass 3: normalize ----------------
__global__ void k_normalize(float* __restrict__ nrm, int n_vtx) {
    int i = blockIdx.x * blockDim.x + threadIdx.x;
    if (i >= n_vtx) return;
    float x = nrm[i * 3 + 0], y = nrm[i * 3 + 1], z = nrm[i * 3 + 2];
    float s = 1.0f / fmaxf(sqrtf(x * x + y * y + z * z), 1e-6f);
    nrm[i * 3 + 0] = x * s;
    nrm[i * 3 + 1] = y * s;
    nrm[i * 3 + 2] = z * s;
}

// ---------------- Pass 4: fused pixel shading with WMMA lighting ----------------
// One wave = 16 pixels; lanes 16..31 mirror pixels 0..15 (fits the f32 WMMA
// A/B layouts: VGPR0 = {K0 | K2}, VGPR1 = {K1 | K3}).
__global__ void __launch_bounds__(256)
k_shade(const float* __restrict__ colors,     // (1,V,3)
        const float* __restrict__ lights,     // (B,L,6)
        const int*   __restrict__ faces,      // (F,3)
        const int*   __restrict__ p2f,        // (B,H,W,1)
        const float* __restrict__ bary,       // (B,H,W,1,3)
        const float* __restrict__ nrm,        // (B,V,3) workspace
        float*       __restrict__ out)        // (B,3,H,W)
{
    const int lane  = threadIdx.x & 31;       // wave32
    const int wave  = (blockIdx.x * blockDim.x + threadIdx.x) >> 5;
    const bool lo   = lane < 16;
    const int  p    = lane & 15;              // pixel-in-tile (mirrored in hi half)
    const int  pix  = wave * 16 + p;          // global pixel id, < B*HW by grid sizing
    const int  b    = pix / HW;               // wave-uniform (HW % 16 == 0)

    // ---- gather + barycentric interpolation (each lane does pixel lane&15) ----
    int face_id = p2f[pix];
    float okf   = face_id >= 0 ? 1.0f : 0.0f; // masked pixels -> zero attributes
    int fid     = face_id >= 0 ? face_id : 0;
    int fb      = fid / Fc;                   // batch whose normals this face uses
    int ff      = fid - fb * Fc;
    int i0 = faces[ff * 3 + 0];
    int i1 = faces[ff * 3 + 1];
    int i2 = faces[ff * 3 + 2];
    float w0 = bary[(size_t)pix * 3 + 0];
    float w1 = bary[(size_t)pix * 3 + 1];
    float w2 = bary[(size_t)pix * 3 + 2];

    const float* nb = nrm + (size_t)fb * Vc * 3;
    float alb[3], nml[3];
#pragma unroll
    for (int c = 0; c < 3; ++c) {
        alb[c] = okf * (w0 * colors[i0 * 3 + c] + w1 * colors[i1 * 3 + c] + w2 * colors[i2 * 3 + c]);
        nml[c] = okf * (w0 * nb[i0 * 3 + c] + w1 * nb[i1 * 3 + c] + w2 * nb[i2 * 3 + c]);
    }

    // ---- lighting setup (per-batch lights; wave-uniform b) ----
    const float* lb = lights + (size_t)b * Lc * 6;
    // Matmul 1: D1[l][p] = d_l . n_p   (A = light dirs 16x4, rows 0..4; B = normals^T 4x16)
    int   l   = lane & 15;
    int   ll  = l < Lc ? l : Lc - 1;
    float lv  = l < Lc ? 1.0f : 0.0f;
    float dx = lb[ll * 6 + 0], dy = lb[ll * 6 + 1], dz = lb[ll * 6 + 2];
    float inv = lv / fmaxf(sqrtf(dx * dx + dy * dy + dz * dz), 1e-12f);

    v2f a1; a1[0] = lo ? dx * inv : dz * inv;  a1[1] = lo ? dy * inv : 0.0f;
    v2f b1; b1[0] = lo ? nml[0]   : nml[2];    b1[1] = lo ? nml[1]   : 0.0f;

    float shade0, shade1, shade2;
#if __has_builtin(__builtin_amdgcn_wmma_f32_16x16x4_f32)
    v8f zc = {};
    v8f d1 = __builtin_amdgcn_wmma_f32_16x16x4_f32(false, a1, false, b1,
                                                   (short)0, zc, false, false);
#pragma unroll
    for (int i = 0; i < 8; ++i) d1[i] = fminf(fmaxf(d1[i], 0.0f), 1.0f);

    // Matmul 2 (K=5 via two accumulating WMMAs): D2[c][p] = sum_l I_l[c] * ndl[l][p]
    int   c  = lane & 15;
    float cv = c < 3 ? 1.0f : 0.0f;
    int   cc = c < 3 ? c : 0;
    float I0 = lb[0 * 6 + 3 + cc], I1 = lb[1 * 6 + 3 + cc];
    float I2 = lb[2 * 6 + 3 + cc], I3 = lb[3 * 6 + 3 + cc];
    float I4 = lb[4 * 6 + 3 + cc];
    v2f a2a; a2a[0] = cv * (lo ? I0 : I2);  a2a[1] = cv * (lo ? I1 : I3);
    v2f a2b; a2b[0] = lo ? cv * I4 : 0.0f;  a2b[1] = 0.0f;

    // B2 rows {0,2} / {1,3}: rows 2,3 of ndl^T live in lanes 0..15 of d1[2]/d1[3]
    float r2 = __shfl(d1[2], lane & 15, 32);
    float r3 = __shfl(d1[3], lane & 15, 32);
    v2f b2a; b2a[0] = lo ? d1[0] : r2;      b2a[1] = lo ? d1[1] : r3;
    v2f b2b; b2b[0] = lo ? d1[4] : 0.0f;    b2b[1] = 0.0f;

    v8f d2 = __builtin_amdgcn_wmma_f32_16x16x4_f32(false, a2a, false, b2a,
                                                   (short)0, zc, false, false);
    d2     = __builtin_amdgcn_wmma_f32_16x16x4_f32(false, a2b, false, b2b,
                                                   (short)0, d2, false, false);
    // D2 rows 0..2 (channels), lanes 0..15, col = pixel = lane
    shade0 = d2[0] * 0.2f;
    shade1 = d2[1] * 0.2f;
    shade2 = d2[2] * 0.2f;
#else
    // Scalar fallback (should not be taken on gfx1250)
    float s0 = 0.f, s1 = 0.f, s2 = 0.f;
    for (int li = 0; li < Lc; ++li) {
        float ex = lb[li * 6 + 0], ey = lb[li * 6 + 1], ez = lb[li * 6 + 2];
        float is = 1.0f / fmaxf(sqrtf(ex * ex + ey * ey + ez * ez), 1e-12f);
        float nd = fminf(fmaxf((nml[0] * ex + nml[1] * ey + nml[2] * ez) * is, 0.f), 1.f);
        s0 += nd * lb[li * 6 + 3];
        s1 += nd * lb[li * 6 + 4];
        s2 += nd * lb[li * 6 + 5];
    }
    shade0 = s0 * 0.2f; shade1 = s1 * 0.2f; shade2 = s2 * 0.2f;
#endif

    if (lo) {
        int rem = pix - b * HW;
        out[((size_t)b * 3 + 0) * HW + rem] = alb[0] * shade0;
        out[((size_t)b * 3 + 1) * HW + rem] = alb[1] * shade1;
        out[((size_t)b * 3 + 2) * HW + rem] = alb[2] * shade2;
    }
}

// ---------------------------------------------------------------------------
extern "C" void kernel_launch(void* const* d_in, const int* in_sizes, int n_in,
                              void* d_out, int out_size, void* d_ws, size_t ws_size,
                              hipStream_t stream) {
    const float* vertices = (const float*)d_in[0];  // (B,V,3)
    const float* colors   = (const float*)d_in[1];  // (1,V,3)
    const float* lights   = (const float*)d_in[2];  // (B,L,6)
    const int*   faces    = (const int*)  d_in[3];  // (F,3)
    const int*   p2f      = (const int*)  d_in[4];  // (B,H,W,1)
    const float* bary     = (const float*)d_in[5];  // (B,H,W,1,3)
    float* out = (float*)d_out;                     // (B,3,H,W)
    float* nrm = (float*)d_ws;                      // (B,V,3) accumulator, 482 KB

    const int nAcc = Bc * Vc * 3;
    k_zero<<<(nAcc + 255) / 256, 256, 0, stream>>>(nrm, nAcc);

    const int nFace = Bc * Fc;
    k_face_accum<<<(nFace + 255) / 256, 256, 0, stream>>>(vertices, faces, nrm);

    const int nVtx = Bc * Vc;
    k_normalize<<<(nVtx + 255) / 256, 256, 0, stream>>>(nrm, nVtx);

    // 16 pixels per wave, 8 waves per block -> 128 pixels/block
    const int nPix   = Bc * HW;                 // 2,097,152 (multiple of 128)
    const int blocks = nPix / 128;              // 16384
    k_shade<<<blocks, 256, 0, stream>>>(colors, lights, faces, p2f, bary, nrm, out);
}